// TCM_77464030151162
// MI455X (gfx1250) — compile-verified
//
#include <hip/hip_runtime.h>

typedef __attribute__((ext_vector_type(2))) float v2f;
typedef __attribute__((ext_vector_type(8))) float v8f;

#define WSTRIDE 66  // padded LDS row stride (floats): lanes hit distinct banks

__device__ __forceinline__ v8f wmma4(v2f a, v2f b, v8f c) {
  // D = A(16x4 f32) * B(4x16 f32) + C(16x16 f32)
  return __builtin_amdgcn_wmma_f32_16x16x4_f32(
      /*neg_a=*/false, a, /*neg_b=*/false, b,
      /*c_mod=*/(short)0, c, /*reuse_a=*/false, /*reuse_b=*/false);
}

// ---------------------------------------------------------------------------
// Kernel 1: y[N,128] = x[N,128] @ w1[128,128] + b1
// one wave -> 16 rows x 128 cols; 8 waves/block.
// w1 staged transposed in LDS in two 64-row halves: sWT[d*66 + c] = w1[c][d].
// ---------------------------------------------------------------------------
__global__ __launch_bounds__(256) void k_gemm_in(
    const float* __restrict__ x, const float* __restrict__ w1,
    const float* __restrict__ b1, float* __restrict__ y, int n) {
  __shared__ float sWT[128 * WSTRIDE];  // 33 KB
  const int wave = threadIdx.x >> 5;
  const int lane = threadIdx.x & 31;
  const int tile = blockIdx.x * 8 + wave;
  const bool active = tile * 16 < n;  // wave-uniform
  const int row0 = tile * 16;
  const int m  = lane & 15;
  const int kh = lane >> 4;
  const float* arow = x + (size_t)(row0 + m) * 128;

  v8f acc[8] = {};

  for (int half = 0; half < 2; ++half) {
    // stage rows [half*64, half*64+64) of w1, transposed
    for (int i = threadIdx.x; i < 2048; i += 256) {
      const int c = i >> 5;          // 0..63 (source row within half)
      const int d = (i & 31) * 4;    // 0..124 (source col)
      float4 v = *(const float4*)(w1 + (size_t)(half * 64 + c) * 128 + d);
      sWT[(d + 0) * WSTRIDE + c] = v.x;
      sWT[(d + 1) * WSTRIDE + c] = v.y;
      sWT[(d + 2) * WSTRIDE + c] = v.z;
      sWT[(d + 3) * WSTRIDE + c] = v.w;
    }
    __syncthreads();

    if (active) {
      v2f areg[16];
#pragma unroll
      for (int i = 0; i < 16; ++i)
        areg[i] = *(const v2f*)(arow + half * 64 + i * 4 + 2 * kh);
#pragma unroll
      for (int i = 0; i < 16; ++i) {
        const int crow = i * 4 + 2 * kh;
#pragma unroll
        for (int t = 0; t < 8; ++t) {
          const int col = t * 16 + m;
          v2f b = *(const v2f*)(sWT + col * WSTRIDE + crow);
          acc[t] = wmma4(areg[i], b, acc[t]);
        }
      }
    }
    __syncthreads();  // all reads done before restaging
  }

  if (active) {
#pragma unroll
    for (int t = 0; t < 8; ++t) {
      const int col = t * 16 + m;
      const float bias = b1[col];
#pragma unroll
      for (int v = 0; v < 8; ++v) {
        const int row = row0 + v + 8 * kh;
        y[(size_t)row * 128 + col] = acc[t][v] + bias;
      }
    }
  }
}

// ---------------------------------------------------------------------------
// Kernels 2/3: out[n,d] = relu( sum_k sum_c feat[nbr[n,k], c] * W[k,c,d] + b[d] )
// Double-buffered transposed W in LDS; next-k W issued before compute, stored
// after compute; next-k neighbor index prefetched one iteration ahead.
// ---------------------------------------------------------------------------
__global__ __launch_bounds__(256) void k_sconv(
    const float* __restrict__ feat, int ld, const int* __restrict__ nbr,
    const float* __restrict__ W, const float* __restrict__ bias,
    float* __restrict__ out, int n) {
  __shared__ float sWT[2][64 * WSTRIDE];  // 2 x 16.9 KB
  const int wave = threadIdx.x >> 5;
  const int lane = threadIdx.x & 31;
  const int tile = blockIdx.x * 8 + wave;
  const bool active = tile * 16 < n;  // wave-uniform
  const int row0 = tile * 16;
  const int m  = lane & 15;
  const int kh = lane >> 4;
  const int* nrow = nbr + (size_t)(row0 + m) * 27;

  v8f acc[4] = {};

  // stage W[0] into buffer 0
  for (int i = threadIdx.x; i < 1024; i += 256) {
    const int c = i >> 4;         // 0..63
    const int d = (i & 15) * 4;   // 0..60
    float4 v = *(const float4*)(W + (size_t)c * 64 + d);
    sWT[0][(d + 0) * WSTRIDE + c] = v.x;
    sWT[0][(d + 1) * WSTRIDE + c] = v.y;
    sWT[0][(d + 2) * WSTRIDE + c] = v.z;
    sWT[0][(d + 3) * WSTRIDE + c] = v.w;
  }
  __syncthreads();

  int idx_cur = active ? nrow[0] : -1;

  for (int k = 0; k < 27; ++k) {
    // prefetch next neighbor index (used next iteration)
    int idx_next = -1;
    if (active && (k + 1 < 27)) idx_next = nrow[k + 1];

    // issue next W tile's global loads into registers (store after compute)
    float4 wbuf[4];
    const bool have_next = (k + 1) < 27;
    if (have_next) {
      const float* src = W + (size_t)(k + 1) * 4096;
#pragma unroll
      for (int j = 0; j < 4; ++j)
        wbuf[j] = *(const float4*)(src + (threadIdx.x + j * 256) * 4);
    }

    if (active) {
      const bool ok = idx_cur >= 0;
      const float* arow = feat + (size_t)(ok ? idx_cur : 0) * ld;
      v2f areg[16];
#pragma unroll
      for (int i = 0; i < 16; ++i) {
        if (ok) {
          areg[i] = *(const v2f*)(arow + i * 4 + 2 * kh);
        } else {
          areg[i].x = 0.f; areg[i].y = 0.f;
        }
      }
      const float* buf = sWT[k & 1];
#pragma unroll
      for (int i = 0; i < 16; ++i) {
        const int crow = i * 4 + 2 * kh;
#pragma unroll
        for (int t = 0; t < 4; ++t) {
          const int col = t * 16 + m;
          v2f b = *(const v2f*)(buf + col * WSTRIDE + crow);
          acc[t] = wmma4(areg[i], b, acc[t]);
        }
      }
    }

    // store next W tile (other buffer; loads retired long ago by now)
    if (have_next) {
      float* dstb = sWT[(k + 1) & 1];
#pragma unroll
      for (int j = 0; j < 4; ++j) {
        const int i = threadIdx.x + j * 256;
        const int c = i >> 4;
        const int d = (i & 15) * 4;
        dstb[(d + 0) * WSTRIDE + c] = wbuf[j].x;
        dstb[(d + 1) * WSTRIDE + c] = wbuf[j].y;
        dstb[(d + 2) * WSTRIDE + c] = wbuf[j].z;
        dstb[(d + 3) * WSTRIDE + c] = wbuf[j].w;
      }
    }
    __syncthreads();
    idx_cur = idx_next;
  }

  if (active) {
#pragma unroll
    for (int t = 0; t < 4; ++t) {
      const int col = t * 16 + m;
      const float bv = bias[col];
#pragma unroll
      for (int v = 0; v < 8; ++v) {
        const int row = row0 + v + 8 * kh;
        const float val = acc[t][v] + bv;
        out[(size_t)row * 64 + col] = val > 0.f ? val : 0.f;
      }
    }
  }
}

// ---------------------------------------------------------------------------
// Kernel 4: out = x + [ (2*y[:, :64] + r2) | y[:, 64:] ] @ w2 + b2
// ---------------------------------------------------------------------------
__global__ __launch_bounds__(256) void k_gemm_out(
    const float* __restrict__ x, const float* __restrict__ y,
    const float* __restrict__ r2, const float* __restrict__ w2,
    const float* __restrict__ b2, float* __restrict__ out, int n) {
  __shared__ float sWT[128 * WSTRIDE];  // 33 KB
  const int wave = threadIdx.x >> 5;
  const int lane = threadIdx.x & 31;
  const int tile = blockIdx.x * 8 + wave;
  const bool active = tile * 16 < n;
  const int row0 = tile * 16;
  const int m  = lane & 15;
  const int kh = lane >> 4;
  const float* yrow = y  + (size_t)(row0 + m) * 128;
  const float* rrow = r2 + (size_t)(row0 + m) * 64;

  v8f acc[8] = {};

  for (int half = 0; half < 2; ++half) {
    for (int i = threadIdx.x; i < 2048; i += 256) {
      const int c = i >> 5;
      const int d = (i & 31) * 4;
      float4 v = *(const float4*)(w2 + (size_t)(half * 64 + c) * 128 + d);
      sWT[(d + 0) * WSTRIDE + c] = v.x;
      sWT[(d + 1) * WSTRIDE + c] = v.y;
      sWT[(d + 2) * WSTRIDE + c] = v.z;
      sWT[(d + 3) * WSTRIDE + c] = v.w;
    }
    __syncthreads();

    if (active) {
      v2f areg[16];
      if (half == 0) {  // conv half: 2*conv_x + r2
#pragma unroll
        for (int i = 0; i < 16; ++i) {
          v2f cy = *(const v2f*)(yrow + i * 4 + 2 * kh);
          v2f rr = *(const v2f*)(rrow + i * 4 + 2 * kh);
          areg[i].x = 2.f * cy.x + rr.x;
          areg[i].y = 2.f * cy.y + rr.y;
        }
      } else {          // trans half passes through
#pragma unroll
        for (int i = 0; i < 16; ++i)
          areg[i] = *(const v2f*)(yrow + 64 + i * 4 + 2 * kh);
      }
#pragma unroll
      for (int i = 0; i < 16; ++i) {
        const int crow = i * 4 + 2 * kh;
#pragma unroll
        for (int t = 0; t < 8; ++t) {
          const int col = t * 16 + m;
          v2f b = *(const v2f*)(sWT + col * WSTRIDE + crow);
          acc[t] = wmma4(areg[i], b, acc[t]);
        }
      }
    }
    __syncthreads();
  }

  if (active) {
#pragma unroll
    for (int t = 0; t < 8; ++t) {
      const int col = t * 16 + m;
      const float bias = b2[col];
#pragma unroll
      for (int v = 0; v < 8; ++v) {
        const int row = row0 + v + 8 * kh;
        const size_t o = (size_t)row * 128 + col;
        out[o] = x[o] + acc[t][v] + bias;
      }
    }
  }
}

// ---------------------------------------------------------------------------
extern "C" void kernel_launch(void* const* d_in, const int* in_sizes, int n_in,
                              void* d_out, int out_size, void* d_ws, size_t ws_size,
                              hipStream_t stream) {
  const float* x   = (const float*)d_in[0];
  const float* w1  = (const float*)d_in[1];
  const float* b1  = (const float*)d_in[2];
  const float* w2  = (const float*)d_in[3];
  const float* b2  = (const float*)d_in[4];
  const float* rw1 = (const float*)d_in[5];
  const float* rb1 = (const float*)d_in[6];
  const float* rw2 = (const float*)d_in[7];
  const float* rb2 = (const float*)d_in[8];
  const int*   nbr = (const int*)d_in[9];

  const int n = in_sizes[0] / 128;  // number of points (100000)

  float* y  = (float*)d_ws;                 // N x 128
  float* r1 = y  + (size_t)n * 128;         // N x 64
  float* r2 = r1 + (size_t)n * 64;          // N x 64

  const int tiles  = (n + 15) / 16;         // 6250, exact
  const int blocks = (tiles + 7) / 8;       // 8 waves (tiles) per 256-thread block

  k_gemm_in <<<blocks, 256, 0, stream>>>(x, w1, b1, y, n);
  k_sconv   <<<blocks, 256, 0, stream>>>(y, 128, nbr, rw1, rb1, r1, n);
  k_sconv   <<<blocks, 256, 0, stream>>>(r1, 64, nbr, rw2, rb2, r2, n);
  k_gemm_out<<<blocks, 256, 0, stream>>>(x, y, r2, w2, b2, (float*)d_out, n);
}